// AttentionReadout2DPDE_89661737271692
// MI455X (gfx1250) — compile-verified
//
#include <hip/hip_runtime.h>
#include <math.h>

typedef float v2f __attribute__((ext_vector_type(2)));
typedef float v4f __attribute__((ext_vector_type(4)));
typedef float v8f __attribute__((ext_vector_type(8)));

#define B_    256
#define S_    256
#define NX_   512
#define NY_   512
#define Q_    64
#define D_    32
#define H_    256
#define RAWC  (2 + D_)        // 34
#define NOUT  (Q_ * RAWC)     // 2176
#define SPLIT 8               // partial-reduction blocks per batch

// ---------------------------------------------------------------------------
// Phase A1: partial sums for per-batch mean/var. 256 MB streamed once ->
// pure bandwidth phase; 2048 blocks of 256 threads guarantees full-device
// occupancy on a large WGP count. Non-temporal B128 loads (data used once).
// ---------------------------------------------------------------------------
__global__ void field_stats_partial_kernel(const float* __restrict__ field,
                                           float* __restrict__ psum,
                                           float* __restrict__ psumsq)
{
    const int b    = blockIdx.x / SPLIT;
    const int part = blockIdx.x - b * SPLIT;
    const int tid  = threadIdx.x;                       // 256 threads
    const int chunk4 = (NX_ * NY_) / 4 / SPLIT;         // 8192 float4 per block

    const v4f* f4 = reinterpret_cast<const v4f*>(field + (size_t)b * NX_ * NY_)
                  + (size_t)part * chunk4;

    float s = 0.f, s2 = 0.f;
    for (int i = tid; i < chunk4; i += 256) {           // 32 B128 loads / thread
        v4f v = __builtin_nontemporal_load(&f4[i]);
        s  += v.x + v.y + v.z + v.w;
        s2 += v.x * v.x + v.y * v.y + v.z * v.z + v.w * v.w;
    }

    __shared__ float ls[256];
    __shared__ float ls2[256];
    ls[tid]  = s;
    ls2[tid] = s2;
    __syncthreads();
    for (int off = 128; off > 0; off >>= 1) {
        if (tid < off) { ls[tid] += ls[tid + off]; ls2[tid] += ls2[tid + off]; }
        __syncthreads();
    }
    if (tid == 0) {
        psum[b * SPLIT + part]   = ls[0];
        psumsq[b * SPLIT + part] = ls2[0];
    }
}

// ---------------------------------------------------------------------------
// Phase A2: fold SPLIT partials per batch into mu / 1/std (ddof=1).
// One block of 256 threads, one batch per thread.
// ---------------------------------------------------------------------------
__global__ void field_stats_finalize_kernel(const float* __restrict__ psum,
                                            const float* __restrict__ psumsq,
                                            float* __restrict__ mu,
                                            float* __restrict__ invsd)
{
    const int b = threadIdx.x;                          // 256 threads = B_
    float sum = 0.f, sumsq = 0.f;
#pragma unroll
    for (int p = 0; p < SPLIT; ++p) {
        sum   += psum[b * SPLIT + p];
        sumsq += psumsq[b * SPLIT + p];
    }
    const float N = (float)(NX_ * NY_);
    float m   = sum / N;
    float var = (sumsq - sum * m) / (N - 1.0f);         // ddof = 1
    var = fmaxf(var, 0.f);
    float sd = fmaxf(sqrtf(var), 1e-6f);
    mu[b]    = m;
    invsd[b] = 1.0f / sd;
}

// ---------------------------------------------------------------------------
// FP32 GEMM with V_WMMA_F32_16X16X4_F32. One wave per 16x16 output tile,
// K advanced 4 per WMMA. A row-major [M,K], B row-major [K,N].
// Per ISA 7.12.2 (32-bit layouts):
//   A 16x4:  lane l(0-15)->M=l, VGPR0/1 = K 0/1; lanes 16-31 -> K 2/3.
//   B 4x16:  VGPR v, lane-half h -> K = v + 2h, N = lane&15.
//   C 16x16: c[i] -> M = i + 8*h, N = lane&15.
// ---------------------------------------------------------------------------
__global__ void gemm_wmma_f32_kernel(const float* __restrict__ A,
                                     const float* __restrict__ Bm,
                                     const float* __restrict__ bias,
                                     float* __restrict__ C,
                                     int M, int N, int K, int relu)
{
    const int lane   = threadIdx.x & 31;
    const int wave   = blockIdx.x * (blockDim.x >> 5) + (threadIdx.x >> 5);
    const int tilesN = N >> 4;
    const int tm     = wave / tilesN;
    const int tn     = wave - tm * tilesN;
    const int h      = lane >> 4;      // lane-half selects K pair
    const int l      = lane & 15;      // M row for A, N col for B/C

    const float* Arow = A  + (size_t)(tm * 16 + l) * K;
    const float* Bcol = Bm + (tn * 16 + l);

    v8f c = {0.f, 0.f, 0.f, 0.f, 0.f, 0.f, 0.f, 0.f};
    for (int k0 = 0; k0 < K; k0 += 4) {
        const int ka = k0 + 2 * h;
        v2f a, bb;
        a.x  = Arow[ka];
        a.y  = Arow[ka + 1];
        bb.x = Bcol[(size_t)ka * N];
        bb.y = Bcol[(size_t)(ka + 1) * N];
        // 8 args: (neg_a, A, neg_b, B, c_mod, C, reuse_a, reuse_b)
        c = __builtin_amdgcn_wmma_f32_16x16x4_f32(
                false, a, false, bb, (short)0, c, false, false);
    }

    const int   col = tn * 16 + l;
    const float bv  = bias[col];
#pragma unroll
    for (int i = 0; i < 8; ++i) {
        int   row = tm * 16 + i + 8 * h;
        float v   = c[i] + bv;
        if (relu) v = fmaxf(v, 0.f);
        C[(size_t)row * N + col] = v;
    }
}

// ---------------------------------------------------------------------------
// Phase C/D: sigmoid -> bilinear sample of normalized field -> readout einsum.
// One 64-thread block per batch. pde_vals staged in LDS.
// ---------------------------------------------------------------------------
__global__ void sample_readout_kernel(const float* __restrict__ raw,
                                      const float* __restrict__ field,
                                      const float* __restrict__ mu,
                                      const float* __restrict__ invsd,
                                      float* __restrict__ out)
{
    const int b   = blockIdx.x;
    const int tid = threadIdx.x;                 // 64 threads
    __shared__ float pde[Q_];
    const float* rawb = raw + (size_t)b * NOUT;

    if (tid < Q_) {
        const int q = tid;
        float rx = rawb[q * RAWC + 0];
        float ry = rawb[q * RAWC + 1];
        float x  = 1.0f / (1.0f + __expf(-rx));  // sigmoid
        float y  = 1.0f / (1.0f + __expf(-ry));
        float px = x * (float)(NY_ - 1);
        float py = y * (float)(NX_ - 1);
        float fx0 = floorf(px), fy0 = floorf(py);
        int x0 = (int)fminf(fmaxf(fx0, 0.f), (float)(NY_ - 1));
        int y0 = (int)fminf(fmaxf(fy0, 0.f), (float)(NX_ - 1));
        int x1 = min(x0 + 1, NY_ - 1);
        int y1 = min(y0 + 1, NX_ - 1);
        float wx = px - (float)x0;
        float wy = py - (float)y0;
        const float* fb = field + (size_t)b * NX_ * NY_;
        float g00 = fb[y0 * NY_ + x0];
        float g01 = fb[y0 * NY_ + x1];
        float g10 = fb[y1 * NY_ + x0];
        float g11 = fb[y1 * NY_ + x1];
        float v = g00 * (1.f - wx) * (1.f - wy) + g01 * wx * (1.f - wy)
                + g10 * (1.f - wx) * wy        + g11 * wx * wy;
        pde[q] = (v - mu[b]) * invsd[b];
    }
    __syncthreads();
    if (tid < D_) {
        const int d = tid;
        float acc = 0.f;
        for (int q = 0; q < Q_; ++q)
            acc = fmaf(pde[q], rawb[q * RAWC + 2 + d], acc);
        out[b * D_ + d] = acc;
    }
}

// ---------------------------------------------------------------------------
extern "C" void kernel_launch(void* const* d_in, const int* in_sizes, int n_in,
                              void* d_out, int out_size, void* d_ws, size_t ws_size,
                              hipStream_t stream)
{
    const float* meas  = (const float*)d_in[0];   // [B,S]
    const float* field = (const float*)d_in[1];   // [B,NX,NY]
    const float* W1    = (const float*)d_in[2];   // [S,H]
    const float* b1    = (const float*)d_in[3];   // [H]
    const float* W2    = (const float*)d_in[4];   // [H,NOUT]
    const float* b2    = (const float*)d_in[5];   // [NOUT]
    float* out = (float*)d_out;                   // [B,D]

    // Workspace (floats):
    // mu[256] | invsd[256] | psum[256*8] | psumsq[256*8] | hid[256*256] | raw[256*2176]
    float* mu     = (float*)d_ws;
    float* invsd  = mu + B_;
    float* psum   = invsd + B_;
    float* psumsq = psum + B_ * SPLIT;
    float* hid    = psumsq + B_ * SPLIT;
    float* raw    = hid + (size_t)B_ * H_;

    // A: per-batch field statistics (bandwidth-dominant phase, 2048 blocks).
    field_stats_partial_kernel<<<B_ * SPLIT, 256, 0, stream>>>(field, psum, psumsq);
    field_stats_finalize_kernel<<<1, B_, 0, stream>>>(psum, psumsq, mu, invsd);

    // B: hid = relu(meas @ W1 + b1). 256 tiles -> 32 blocks x 8 waves.
    gemm_wmma_f32_kernel<<<(B_ / 16) * (H_ / 16) / 8, 256, 0, stream>>>(
        meas, W1, b1, hid, B_, H_, S_, 1);

    // C: raw = hid @ W2 + b2. 16*136 = 2176 tiles -> 272 blocks x 8 waves.
    gemm_wmma_f32_kernel<<<(B_ / 16) * (NOUT / 16) / 8, 256, 0, stream>>>(
        hid, W2, b2, raw, B_, NOUT, H_, 0);

    // D: sigmoid + bilinear sample + readout einsum.
    sample_readout_kernel<<<B_, 64, 0, stream>>>(raw, field, mu, invsd, out);
}